// CustomQuantizer_2345052144227
// MI455X (gfx1250) — compile-verified
//
#include <hip/hip_runtime.h>

// Problem constants (from the reference):
//   x   : [8192, 8192] f32   (N_TOKENS x QUANT_DIM)
//   W   : [1024, 8192] f32   (OUTPUT_DIM x QUANT_DIM)
//   out : [8192, 1024] f32   out[n][j] = W[j][argmax_c x[n][c]]
#define N_TOKENS   8192
#define QUANT_DIM  8192
#define OUTPUT_DIM 1024

// One block per token row. 256 threads = 8 wave32s.
// Each wave streams its 1/8 of the row through a private double-buffered LDS
// stage filled by global_load_async_to_lds_b128 (gfx1250 async DMA engine,
// tracked with ASYNCcnt). The s_wait_dscnt 0 inside the issue asm closes the
// WAR hazard between the previous ds_load_b128 readback of a stage and the
// async engine overwriting that stage (DScnt and ASYNCcnt are unordered
// relative to each other per the ISA).
__global__ __launch_bounds__(256) void quantizer_argmax_gather(
    const float* __restrict__ x,
    const float* __restrict__ W,
    float* __restrict__ out)
{
    // Async staging: 8 waves * 2 stages * 32 lanes * 16B = 8 KiB
    __shared__ __align__(16) unsigned char s_stage[8192];
    __shared__ float s_wv[8];
    __shared__ int   s_wi[8];
    __shared__ int   s_win;

    const int row  = blockIdx.x;
    const int tid  = threadIdx.x;
    const int wave = tid >> 5;   // 0..7
    const int lane = tid & 31;   // 0..31

    const float* rowp = x + (size_t)row * QUANT_DIM;

    float bv = -__builtin_inff();
    int   bi = 0;

#if defined(__gfx1250__)
    // Low 32 bits of the generic shared-pointer are the LDS byte offset
    // (flat/LDS aperture truncation rule), which is what the async-to-LDS
    // VDST operand expects.
    const unsigned lds_base = (unsigned)(size_t)&s_stage[0];

    // Prime the pipe: chunk 0 for this wave. (DScnt is 0 here; the wait is
    // harmless and keeps the asm template uniform.)
    {
        unsigned ldsa = lds_base + (unsigned)((wave * 2 + 0) * 512 + lane * 16);
        unsigned long long ga =
            (unsigned long long)(rowp + 0 * 1024 + wave * 128 + lane * 4);
        asm volatile("s_wait_dscnt 0x0\n\t"
                     "global_load_async_to_lds_b128 %0, %1, off"
                     :: "v"(ldsa), "v"(ga) : "memory");
    }

    #pragma unroll
    for (int i = 0; i < 8; ++i) {
        if (i + 1 < 8) {
            // Ensure the ds_load readback of the stage we are about to
            // overwrite has completed (s_wait_dscnt 0), then prefetch the
            // next chunk, then wait for the oldest async op: ASYNCcnt <= 1.
            unsigned ldsa = lds_base +
                (unsigned)((wave * 2 + ((i + 1) & 1)) * 512 + lane * 16);
            unsigned long long ga =
                (unsigned long long)(rowp + (i + 1) * 1024 + wave * 128 + lane * 4);
            asm volatile("s_wait_dscnt 0x0\n\t"
                         "global_load_async_to_lds_b128 %0, %1, off\n\t"
                         "s_wait_asynccnt 0x1"
                         :: "v"(ldsa), "v"(ga) : "memory");
        } else {
            asm volatile("s_wait_asynccnt 0x0" ::: "memory");
        }

        const float4 v = *(const float4*)
            &s_stage[(wave * 2 + (i & 1)) * 512 + lane * 16];
        const int g = i * 1024 + wave * 128 + lane * 4;
        // Strictly increasing index order per lane + strict '>' keeps the
        // first occurrence on ties, matching jnp.argmax.
        if (v.x > bv) { bv = v.x; bi = g;     }
        if (v.y > bv) { bv = v.y; bi = g + 1; }
        if (v.z > bv) { bv = v.z; bi = g + 2; }
        if (v.w > bv) { bv = v.w; bi = g + 3; }
    }
#else
    // Host pass / non-gfx1250 fallback: direct 128-bit global loads.
    for (int i = 0; i < 8; ++i) {
        const float4 v = *(const float4*)(rowp + i * 1024 + wave * 128 + lane * 4);
        const int g = i * 1024 + wave * 128 + lane * 4;
        if (v.x > bv) { bv = v.x; bi = g;     }
        if (v.y > bv) { bv = v.y; bi = g + 1; }
        if (v.z > bv) { bv = v.z; bi = g + 2; }
        if (v.w > bv) { bv = v.w; bi = g + 3; }
    }
#endif

    // ---- wave32 argmax reduce (lowest index wins ties) ----
    #pragma unroll
    for (int m = 16; m > 0; m >>= 1) {
        float ov = __shfl_xor(bv, m, 32);
        int   oi = __shfl_xor(bi, m, 32);
        if (ov > bv || (ov == bv && oi < bi)) { bv = ov; bi = oi; }
    }
    if (lane == 0) { s_wv[wave] = bv; s_wi[wave] = bi; }
    __syncthreads();

    // ---- cross-wave reduce (8 candidates) ----
    if (tid == 0) {
        float v = s_wv[0];
        int   c = s_wi[0];
        #pragma unroll
        for (int w = 1; w < 8; ++w) {
            if (s_wv[w] > v || (s_wv[w] == v && s_wi[w] < c)) {
                v = s_wv[w]; c = s_wi[w];
            }
        }
        s_win = c;
    }
    __syncthreads();

    const int c = s_win;

    // ---- gather column c of W, coalesced store of out row ----
    // W is [OUTPUT_DIM, QUANT_DIM] row-major; out[n][j] = W[j*QUANT_DIM + c].
    // Reads are 32KB-strided but W (32 MiB) is L2-resident (192 MB L2).
    float* orow = out + (size_t)row * OUTPUT_DIM;
    #pragma unroll
    for (int j = tid; j < OUTPUT_DIM; j += 256) {
        orow[j] = W[(size_t)j * QUANT_DIM + c];
    }
}

extern "C" void kernel_launch(void* const* d_in, const int* in_sizes, int n_in,
                              void* d_out, int out_size, void* d_ws, size_t ws_size,
                              hipStream_t stream) {
    (void)in_sizes; (void)n_in; (void)out_size; (void)d_ws; (void)ws_size;
    const float* x = (const float*)d_in[0];
    const float* W = (const float*)d_in[1];
    float* out = (float*)d_out;
    quantizer_argmax_gather<<<dim3(N_TOKENS), dim3(256), 0, stream>>>(x, W, out);
}